// VectorQuantizer_25271587569752
// MI455X (gfx1250) — compile-verified
//
#include <hip/hip_runtime.h>
#include <hip/hip_bf16.h>
#include <float.h>

// Vector quantization (nearest codebook entry) fused kernels for gfx1250.
//   z:        [8192, 32] fp32   (8*1024 rows)
//   codebook: [16384, 32] fp32  (rows ~unit norm)
//   out:      [8192*32] z_q floats  ++  [8192] indices-as-float
//   ws:       cc[16384] fp32 = per-row squared norms of the codebook
//
// Pass 1: cc[n] = sum_k codebook[n][k]^2  (64 KB, L2-resident, computed once
//         per launch instead of 512x redundantly inside the main loop).
// Pass 2: S = normalize(z) . codebook^T via V_WMMA_F32_16X16X4_F32 (exact
//         fp32 -> argmin matches the fp32 reference), fused streaming argmin
//         so the 512 MB distance matrix never touches memory, then gather.

typedef __attribute__((ext_vector_type(2))) float v2f;
typedef __attribute__((ext_vector_type(4))) float v4f;
typedef __attribute__((ext_vector_type(8))) float v8f;

#define N_EMBED      16384
#define CODE_DIM     32
#define ROWS_TOTAL   8192
#define WAVES_PER_BLOCK 4
#define BLOCK_THREADS   128
#define N_TILES      (N_EMBED / 16)

// ---------------- Pass 1: codebook row squared norms ----------------
__global__ __launch_bounds__(256)
void vq_cc_kernel(const float* __restrict__ codebook, float* __restrict__ cc) {
  const int n = blockIdx.x * 256 + threadIdx.x;   // 16384 rows total
  const float* row = codebook + n * CODE_DIM;
  float s = 0.0f;
  #pragma unroll
  for (int q = 0; q < CODE_DIM / 4; ++q) {
    v4f v = *(const v4f*)(row + 4 * q);
    s += v.x * v.x + v.y * v.y + v.z * v.z + v.w * v.w;
  }
  cc[n] = s;
}

// ---------------- Pass 2: WMMA scores + streaming argmin + gather ----------------
__global__ __launch_bounds__(BLOCK_THREADS)
void vq_wmma_argmin_kernel(const float* __restrict__ z,
                           const float* __restrict__ codebook,
                           const float* __restrict__ ccbuf,
                           float* __restrict__ out) {
  __shared__ float lds_d[WAVES_PER_BLOCK * 16];
  __shared__ int   lds_i[WAVES_PER_BLOCK * 16];
  __shared__ int   fidx[16];

  const int tid  = threadIdx.x;
  const int wave = tid >> 5;          // 4 waves per block (wave32)
  const int lane = tid & 31;
  const int half = lane >> 4;         // 0: lanes 0-15, 1: lanes 16-31
  const int l16  = lane & 15;
  const int m_base = blockIdx.x * 16; // 16 query rows per block

  // ---- Load A fragments (16 rows x K=32 fp32) and L2-normalize in-register.
  // WMMA f32 16x16x4 A layout: lane holds M=lane%16, K = 4*q + 2*(lane/16)+{0,1}
  // Lanes L and L+16 together hold the full row -> one shfl_xor gives ||z||^2.
  v2f a[8];
  float ss = 0.0f;
  {
    const float* zrow = z + (m_base + l16) * CODE_DIM;
    #pragma unroll
    for (int q = 0; q < 8; ++q) {
      v2f v = *(const v2f*)(zrow + 4 * q + 2 * half);
      a[q] = v;
      ss += v.x * v.x + v.y * v.y;
    }
  }
  ss += __shfl_xor(ss, 16, 32);
  const float inv = 1.0f / fmaxf(sqrtf(ss), 1e-12f);
  #pragma unroll
  for (int q = 0; q < 8; ++q) { a[q].x *= inv; a[q].y *= inv; }

  // Running (min distance, index) per C/D accumulator register.
  float bestd[8];
  int   besti[8];
  #pragma unroll
  for (int r = 0; r < 8; ++r) { bestd[r] = FLT_MAX; besti[r] = 0; }

  // ---- Stream codebook tiles of 16 columns; waves stride the N dimension.
  for (int nt = wave; nt < N_TILES; nt += WAVES_PER_BLOCK) {
    const int n = nt * 16 + l16;                 // this lane's column index
    const float* crow = codebook + n * CODE_DIM;
    __builtin_prefetch(crow + WAVES_PER_BLOCK * 16 * CODE_DIM, 0, 3);

    // Precomputed ||c_n||^2 : one scalar load, L2-resident.
    const float cc = ccbuf[n];

    // B layout mirrors A: lane holds N=lane%16, K = 4*q + 2*half + {0,1}.
    v2f bf[8];
    #pragma unroll
    for (int q = 0; q < 8; ++q) {
      bf[q] = *(const v2f*)(crow + 4 * q + 2 * half);
    }

    // 16x16 fp32 score tile: 8 x V_WMMA_F32_16X16X4_F32 covering K=32.
    v8f acc = {};
    #pragma unroll
    for (int q = 0; q < 8; ++q) {
      acc = __builtin_amdgcn_wmma_f32_16x16x4_f32(
          false, a[q], false, bf[q], (short)0, acc, false, false);
    }

    // d_n = ||c_n||^2 - 2 * <zn, c_n>   (||zn||^2 constant per row -> dropped)
    #pragma unroll
    for (int r = 0; r < 8; ++r) {
      const float d = __builtin_fmaf(-2.0f, acc[r], cc);
      if (d < bestd[r]) { bestd[r] = d; besti[r] = n; }  // strict keeps lowest n
    }
  }

  // ---- Argmin across the 16 lanes of each half (columns live across lanes).
  #pragma unroll
  for (int r = 0; r < 8; ++r) {
    float d = bestd[r];
    int   i = besti[r];
    #pragma unroll
    for (int m = 8; m >= 1; m >>= 1) {
      const float od = __shfl_xor(d, m, 16);
      const int   oi = __shfl_xor(i, m, 16);
      if (od < d || (od == d && oi < i)) { d = od; i = oi; }
    }
    bestd[r] = d;
    besti[r] = i;
  }

  // lane 0 holds rows 0..7 of the tile, lane 16 holds rows 8..15.
  if (l16 == 0) {
    #pragma unroll
    for (int r = 0; r < 8; ++r) {
      const int row = r + 8 * half;
      lds_d[wave * 16 + row] = bestd[r];
      lds_i[wave * 16 + row] = besti[r];
    }
  }
  __syncthreads();

  // ---- Combine the 4 waves' candidates per row; write indices.
  if (tid < 16) {
    float d = lds_d[tid];
    int   i = lds_i[tid];
    #pragma unroll
    for (int w = 1; w < WAVES_PER_BLOCK; ++w) {
      const float od = lds_d[w * 16 + tid];
      const int   oi = lds_i[w * 16 + tid];
      if (od < d || (od == d && oi < i)) { d = od; i = oi; }
    }
    fidx[tid] = i;
    out[ROWS_TOTAL * CODE_DIM + m_base + tid] = (float)i;  // indices as fp32
  }
  __syncthreads();

  // ---- Gather z_q = codebook[idx] for the block's 16 rows.
  for (int e = tid; e < 16 * CODE_DIM; e += BLOCK_THREADS) {
    const int row = e >> 5;
    const int col = e & 31;
    out[(m_base + row) * CODE_DIM + col] = codebook[fidx[row] * CODE_DIM + col];
  }
}

extern "C" void kernel_launch(void* const* d_in, const int* in_sizes, int n_in,
                              void* d_out, int out_size, void* d_ws, size_t ws_size,
                              hipStream_t stream) {
  (void)in_sizes; (void)n_in; (void)out_size; (void)ws_size;
  const float* z        = (const float*)d_in[0];   // [8192, 32]
  const float* codebook = (const float*)d_in[1];   // [16384, 32]
  float* cc             = (float*)d_ws;            // 16384 floats (64 KB)
  float* out            = (float*)d_out;           // 8192*32 z_q ++ 8192 idx

  vq_cc_kernel<<<N_EMBED / 256, 256, 0, stream>>>(codebook, cc);

  const int blocks = ROWS_TOTAL / 16;              // 512 blocks, 4 waves each
  vq_wmma_argmin_kernel<<<blocks, BLOCK_THREADS, 0, stream>>>(z, codebook, cc, out);
}